// SLAYFeatures_30245159698782
// MI455X (gfx1250) — compile-verified
//
#include <hip/hip_runtime.h>
#include <math.h>

typedef __attribute__((ext_vector_type(2))) float v2f;
typedef __attribute__((ext_vector_type(4))) float v4f;
typedef __attribute__((ext_vector_type(8))) float v8f;

#define NUM_HEADS 16
#define HEAD_DIM  64
#define POLY_DIM  16
#define NUM_FEAT  32
#define NUM_QUAD  2
#define XSTR      68      // padded LDS row stride (floats): (4*row + k) % 64 distinct -> no bank conflicts
#define WAVES     4

#define OUT_PER_B 16384   // R*H*P*M
#define OUT_PER_R 8192    // H*P*M
#define OUT_PER_H 512     // P*M

__global__ __launch_bounds__(128) void slay_features_kernel(
    const float* __restrict__ x, const float* __restrict__ omega,
    const float* __restrict__ anchors, const float* __restrict__ qn,
    const float* __restrict__ qw, float* __restrict__ out)
{
  __shared__ __align__(16) float lds_x[WAVES][16 * XSTR];
  __shared__ __align__(16) float lds_anch[16 * XSTR];
  __shared__ __align__(16) float lds_poly[WAVES][16 * 17];
  __shared__ __align__(16) float lds_prf[WAVES][2 * 16 * NUM_FEAT];

  const int lane  = threadIdx.x & 31;
  const int wave  = threadIdx.x >> 5;
  const int btile = blockIdx.x >> 2;
  const int head  = ((blockIdx.x & 3) << 2) | wave;
  const int b0    = btile << 4;

  // ---- cooperative anchors (16 x 64) load into padded LDS ----
  {
    const int t = threadIdx.x;
#pragma unroll
    for (int i = 0; i < 2; ++i) {
      int v = i * 128 + t;                 // 256 float4s total
      int row = v >> 4, c4 = (v & 15) << 2;
      v4f val = *(const v4f*)(anchors + row * HEAD_DIM + c4);
      *(v4f*)&lds_anch[row * XSTR + c4] = val;
    }
  }
  // ---- per-wave x tile (16 rows x this head's 64 cols), coalesced ----
  {
#pragma unroll
    for (int i = 0; i < 8; ++i) {
      int v = i * 32 + lane;
      int row = v >> 4, c4 = (v & 15) << 2;
      v4f val = *(const v4f*)(x + (size_t)(b0 + row) * 1024 + head * HEAD_DIM + c4);
      *(v4f*)&lds_x[wave][row * XSTR + c4] = val;
    }
  }
  __syncthreads();

  // ---- per-row inverse norm: inv = 1/max(||x_row||, 1e-6); folded in after WMMA ----
  float acc = 0.0f;
#pragma unroll 4
  for (int k4 = 0; k4 < 16; ++k4) {
    v4f f = *(const v4f*)&lds_x[wave][(lane & 15) * XSTR + (k4 << 2)];
    acc += f.x * f.x + f.y * f.y + f.z * f.z + f.w * f.w;
  }
  const float inv = 1.0f / fmaxf(sqrtf(acc), 1e-6f);
  // D-tile VGPR j holds row j (lanes 0-15) / row j+8 (lanes 16-31)
  float rn[8];
#pragma unroll
  for (int j = 0; j < 8; ++j)
    rn[j] = __shfl(inv, j + ((lane >> 4) << 3), 32);

  // A-operand addressing for V_WMMA_F32_16X16X4_F32:
  // lane holds M = lane&15; VGPR0 = K0 (lo lanes) / K2 (hi lanes); VGPR1 = K1 / K3
  const int aoff = (lane >> 4) << 1;       // 0 or 2
  const int arow = (lane & 15) * XSTR + aoff;
  const int hb   = (lane >> 4) << 3;       // row offset for D halves

  // ---- poly: D[b,p] = sum_k x[b,k] * anchors[p,k]  (B[k][p] = anchors[p][k]) ----
  {
    v8f pacc = {};
#pragma unroll 4
    for (int s = 0; s < 16; ++s) {
      v2f av = *(const v2f*)&lds_x[wave][arow + (s << 2)];
      v2f bv = *(const v2f*)&lds_anch[arow + (s << 2)];   // same (row,k) indexing
      pacc = __builtin_amdgcn_wmma_f32_16x16x4_f32(false, av, false, bv,
                                                   (short)0, pacc, false, false);
    }
#pragma unroll
    for (int j = 0; j < 8; ++j) {
      float pr = pacc[j] * rn[j];
      lds_poly[wave][(j + hb) * 17 + (lane & 15)] = pr * pr * 0.25f; // /sqrt(16)
    }
  }

  // ---- PRF: 2 quad x 2 N-halves, B streamed from L2-resident omega ----
  const float* wb = omega + (size_t)head * HEAD_DIM * NUM_FEAT + aoff * NUM_FEAT + (lane & 15);
#pragma unroll
  for (int r = 0; r < NUM_QUAD; ++r) {
    const float sv = qn[r];
    const float s2 = sqrtf(2.0f * fmaxf(sv, 0.0f));
    const float cr = sqrtf(fmaxf(qw[r], 0.0f)) * 0.17677669529663687f; // 1/sqrt(32)
    const float* wr = wb + (size_t)r * NUM_HEADS * HEAD_DIM * NUM_FEAT;
#pragma unroll
    for (int nh = 0; nh < 2; ++nh) {
      const float* wn = wr + (nh << 4);
      v8f facc = {};
#pragma unroll 4
      for (int s = 0; s < 16; ++s) {
        v2f av = *(const v2f*)&lds_x[wave][arow + (s << 2)];
        v2f bv;
        bv.x = wn[(size_t)(s << 2) * NUM_FEAT];
        bv.y = wn[(size_t)((s << 2) + 1) * NUM_FEAT];
        facc = __builtin_amdgcn_wmma_f32_16x16x4_f32(false, av, false, bv,
                                                     (short)0, facc, false, false);
      }
#pragma unroll
      for (int j = 0; j < 8; ++j) {
        float pr = facc[j] * rn[j];
        float e = __expf(fminf(fmaxf(pr * s2 - sv, -10.0f), 10.0f)) * cr;
        lds_prf[wave][((r << 4) + j + hb) * NUM_FEAT + (nh << 4) + (lane & 15)] = e;
      }
    }
  }

  // ---- fusion + streaming NT stores: out[b, r, h, p, m] = poly[b,p] * prf[r,b,m] ----
  // per (b,r): 512 contiguous floats; wave stores 512B-coalesced b128 groups
  float* ob = out + (size_t)b0 * OUT_PER_B + head * OUT_PER_H;
  const int m4 = (lane & 7) << 2;
  const int pl = lane >> 3;
#pragma unroll 2
  for (int b = 0; b < 16; ++b) {
    float* obb = ob + (size_t)b * OUT_PER_B;
#pragma unroll
    for (int r = 0; r < NUM_QUAD; ++r) {
      v4f pr4 = *(const v4f*)&lds_prf[wave][((r << 4) + b) * NUM_FEAT + m4];
      float* dst = obb + r * OUT_PER_R + lane * 4;
#pragma unroll
      for (int it = 0; it < 4; ++it) {
        float pf = lds_poly[wave][b * 17 + (it << 2) + pl];
        v4f v = pr4 * pf;
        __builtin_nontemporal_store(v, (v4f*)(dst + it * 128));
      }
    }
  }
}

extern "C" void kernel_launch(void* const* d_in, const int* in_sizes, int n_in,
                              void* d_out, int out_size, void* d_ws, size_t ws_size,
                              hipStream_t stream) {
  const float* x       = (const float*)d_in[0];
  const float* omega   = (const float*)d_in[1];
  const float* anchors = (const float*)d_in[2];
  const float* qn      = (const float*)d_in[3];
  const float* qw      = (const float*)d_in[4];
  float* out = (float*)d_out;

  const int B = in_sizes[0] / 1024;        // 8192
  const int grid = (B / 16) * 4;           // 4 head-groups per b-tile, 4 waves each
  slay_features_kernel<<<grid, 128, 0, stream>>>(x, omega, anchors, qn, qw, out);
}